// ODESolver_node_6296422056471
// MI455X (gfx1250) — compile-verified
//
#include <hip/hip_runtime.h>
#include <math.h>

typedef __attribute__((ext_vector_type(2))) float v2f;
typedef __attribute__((ext_vector_type(4))) float v4f;
typedef __attribute__((ext_vector_type(8))) float v8f;

#define N_ROWS 16384
#define DIM    256
#define MT     16          // rows per block
#define SXP    260         // padded LDS row stride (floats) -> conflict-free A reads
#define NT     256         // threads per block (8 waves, wave32)
#define NBLK   (N_ROWS / MT)   // 1024 blocks
#define NITER  16

#if __has_builtin(__builtin_amdgcn_tanhf)
__device__ __forceinline__ float fast_tanh(float x) { return __builtin_amdgcn_tanhf(x); }
#else
__device__ __forceinline__ float fast_tanh(float x) { return tanhf(x); }
#endif

struct Ctrl { float s; float h; int commit; int pad; };

// ---------------------------------------------------------------- init ----
__global__ __launch_bounds__(NT, 1)
void node_init_kernel(const float* __restrict__ r, float* __restrict__ z, Ctrl* c) {
    int i = blockIdx.x * NT + threadIdx.x;
    if (i < N_ROWS * DIM / 4) ((v4f*)z)[i] = ((const v4f*)r)[i];
    if (i == 0) { c->s = 0.0f; c->h = 0.05f; c->commit = 0; }
}

// ------------------------------------------------------------- pack W ----
// Wp[p][n] = (W[2p][n], W[2p+1][n])  ->  one coalesced b64 per B fragment.
__global__ __launch_bounds__(NT, 1)
void node_pack_w_kernel(const float* __restrict__ W, float* __restrict__ Wp) {
    int idx = blockIdx.x * NT + threadIdx.x;         // DIM*DIM = 65536
    int k = idx >> 8, n = idx & (DIM - 1);
    Wp[(size_t)(k >> 1) * (2 * DIM) + n * 2 + (k & 1)] = W[idx];
}

// --------------------------------------------------------- fused stage ----
// One dopri5 step for MT rows: builds all 7 stage inputs in LDS, runs the
// (MTx256)@(256x256) GEMM per stage with V_WMMA_F32_16X16X4_F32, keeps the
// six needed k_j in LDS, emits z_new (FSAL) + deterministic (err/tol)^2 partial.
__global__ __launch_bounds__(NT, 1)
void node_dopri_stage_kernel(const float* __restrict__ z,
                             const float* __restrict__ startT,
                             const float* __restrict__ endT,
                             const float* __restrict__ Wt,
                             const float* __restrict__ bt,
                             const float* __restrict__ Wp,   // packed pair-interleaved W
                             const float* __restrict__ bias,
                             float* __restrict__ zprop,
                             float* __restrict__ partial,
                             const Ctrl* __restrict__ ctrl) {
    __shared__ float sZ[MT * SXP];
    __shared__ float sX[MT * SXP];
    __shared__ float sK[6][MT * SXP];   // k0..k5 (k6 in regs; z_new parked in sK[1] for stage 6)
    __shared__ float sWt[DIM], sBt[DIM], sB[DIM];
    __shared__ float sRatio[MT], sStart[MT];
    __shared__ float sRed[NT];

    const int tid  = threadIdx.x;
    const int lane = tid & 31;
    const int wav  = tid >> 5;
    const int r0   = blockIdx.x * MT;

    sWt[tid] = Wt[tid];
    sBt[tid] = bt[tid];
    sB[tid]  = bias[tid];
    if (tid < MT) {
        float st = startT[r0 + tid];
        sStart[tid] = st;
        sRatio[tid] = endT[r0 + tid] - st;
    }
    const float s     = ctrl->s;
    const float h     = ctrl->h;
    const float h_eff = fminf(h, 1.0f - s);

    #pragma unroll
    for (int rr = 0; rr < MT; ++rr)
        sZ[rr * SXP + tid] = z[(size_t)(r0 + rr) * DIM + tid];
    __syncthreads();

    constexpr double AC[7][6] = {
        {0,0,0,0,0,0},
        {1.0/5,0,0,0,0,0},
        {3.0/40, 9.0/40,0,0,0,0},
        {44.0/45, -56.0/15, 32.0/9,0,0,0},
        {19372.0/6561, -25360.0/2187, 64448.0/6561, -212.0/729,0,0},
        {9017.0/3168, -355.0/33, 46732.0/5247, 49.0/176, -5103.0/18656,0},
        {35.0/384, 0.0, 500.0/1113, 125.0/192, -2187.0/6784, 11.0/84}};
    constexpr double CC[7] = {0.0, 1.0/5, 3.0/10, 4.0/5, 8.0/9, 1.0, 1.0};
    constexpr float E0 = (float)(35.0/384.0     - 5179.0/57600.0);
    constexpr float E2 = (float)(500.0/1113.0   - 7571.0/16695.0);
    constexpr float E3 = (float)(125.0/192.0    - 393.0/640.0);
    constexpr float E4 = (float)(-2187.0/6784.0 + 92097.0/339200.0);
    constexpr float E5 = (float)(11.0/84.0      - 187.0/2100.0);
    constexpr float E6 = (float)(-1.0/40.0);

    const int hi  = lane >> 4;       // K-pair select / C row-half select
    const int l15 = lane & 15;
    const int n0  = wav * 16;        // first col tile for this wave
    const int n1  = n0 + 128;        // second col tile

    // stage-invariant epilogue operands (hoisted LDS reads)
    const float be0 = sB[n0 + l15];
    const float be1 = sB[n1 + l15];
    float rav[8];
    #pragma unroll
    for (int v = 0; v < 8; ++v) rav[v] = sRatio[v + (hi << 3)];

    const v2f* WpV   = (const v2f*)Wp;
    const int  pbase = hi * DIM + n0 + l15;   // v2f index at k=0, tile 0

    float errAcc = 0.0f;

    #pragma unroll
    for (int st = 0; st < 7; ++st) {
        // ---- build X = zi + t*Wt + bt ; (stage 6: zi core IS z_new, FSAL) ----
        const float si = s + (float)CC[st] * h_eff;
        #pragma unroll
        for (int rr = 0; rr < MT; ++rr) {
            const int o = rr * SXP + tid;
            float acc = sZ[o];
            #pragma unroll
            for (int j = 0; j < 6; ++j) {
                if ((float)AC[st][j] != 0.0f)
                    acc += (h_eff * (float)AC[st][j]) * sK[j][o];
            }
            if (st == 6) {
                sK[1][o] = acc;                                   // z_new (k1 dead by now)
                zprop[(size_t)(r0 + rr) * DIM + tid] = acc;       // proposed z
            }
            const float t = si * sRatio[rr] + sStart[rr];
            sX[o] = acc + t * sWt[tid] + sBt[tid];
        }
        __syncthreads();

        // ---- GEMM: X(16x256) @ W(256x256), this wave: col tiles n0, n1 ----
        v8f c0 = {0.f,0.f,0.f,0.f,0.f,0.f,0.f,0.f};
        v8f c1 = {0.f,0.f,0.f,0.f,0.f,0.f,0.f,0.f};
        #pragma unroll 8
        for (int k = 0; k < DIM; k += 4) {
            const int ka = k + (hi << 1);
            const v2f a  = *(const v2f*)(sX + l15 * SXP + ka);          // ds b64
            const int pi = (k >> 1) * DIM + pbase;
            const v2f b0 = WpV[pi];                                     // global b64, coalesced
            const v2f b1 = WpV[pi + 128];
            c0 = __builtin_amdgcn_wmma_f32_16x16x4_f32(false, a, false, b0, (short)0, c0, false, false);
            c1 = __builtin_amdgcn_wmma_f32_16x16x4_f32(false, a, false, b1, (short)0, c1, false, false);
        }

        // ---- epilogue: +bias, tanh (V_TANH_F32), *ratio ----
        if (st < 6) {
            #pragma unroll
            for (int v = 0; v < 8; ++v) {
                const int rr = v + (hi << 3);
                sK[st][rr * SXP + n0 + l15] = fast_tanh(c0[v] + be0) * rav[v];
                sK[st][rr * SXP + n1 + l15] = fast_tanh(c1[v] + be1) * rav[v];
            }
        } else {
            #pragma unroll
            for (int v = 0; v < 8; ++v) {
                const int rr = v + (hi << 3);
                #pragma unroll
                for (int tcol = 0; tcol < 2; ++tcol) {
                    const int col  = (tcol == 0 ? n0 : n1) + l15;
                    const float cv = (tcol == 0 ? c0[v] : c1[v]);
                    const float bb = (tcol == 0 ? be0 : be1);
                    const float k6 = fast_tanh(cv + bb) * rav[v];
                    const int o = rr * SXP + col;
                    const float err = h_eff * (E0 * sK[0][o] + E2 * sK[2][o] + E3 * sK[3][o] +
                                               E4 * sK[4][o] + E5 * sK[5][o] + E6 * k6);
                    const float zn  = sK[1][o];            // z_new parked here
                    const float zo  = sZ[o];
                    const float tol = 1e-7f + 1e-6f * fmaxf(fabsf(zo), fabsf(zn));
                    const float q   = err / tol;
                    errAcc += q * q;
                }
            }
        }
        __syncthreads();
    }

    // deterministic block reduction of sum((err/tol)^2)
    sRed[tid] = errAcc;
    __syncthreads();
    #pragma unroll
    for (int off = NT / 2; off > 0; off >>= 1) {
        if (tid < off) sRed[tid] += sRed[tid + off];
        __syncthreads();
    }
    if (tid == 0) partial[blockIdx.x] = sRed[0];
}

// -------------------------------------------------------------- control ----
__global__ void node_control_kernel(Ctrl* c, const float* __restrict__ partial) {
    if (threadIdx.x != 0 || blockIdx.x != 0) return;
    float sum = 0.0f;
    for (int i = 0; i < NBLK; ++i) sum += partial[i];      // fixed order: deterministic
    const float en     = sqrtf(sum / (float)(N_ROWS * DIM));
    const bool  done   = c->s >= 1.0f - 1e-10f;
    const float h_eff  = fminf(c->h, 1.0f - c->s);
    const bool  accept = en <= 1.0f;
    float factor = 0.9f * powf(en + 1e-16f, -0.2f);
    factor = fminf(fmaxf(factor, 0.2f), 10.0f);
    int commit = 0;
    if (!done) {
        if (accept) { c->s += h_eff; commit = 1; }
        c->h = h_eff * factor;
    }
    c->commit = commit;
}

// --------------------------------------------------------------- commit ----
__global__ __launch_bounds__(NT, 1)
void node_commit_kernel(float* __restrict__ z, const float* __restrict__ zprop,
                        const Ctrl* __restrict__ c) {
    if (!c->commit) return;
    int i = blockIdx.x * NT + threadIdx.x;
    if (i < N_ROWS * DIM / 4) ((v4f*)z)[i] = ((const v4f*)zprop)[i];
}

// --------------------------------------------------------------- launch ----
extern "C" void kernel_launch(void* const* d_in, const int* in_sizes, int n_in,
                              void* d_out, int out_size, void* d_ws, size_t ws_size,
                              hipStream_t stream) {
    const float* r      = (const float*)d_in[0];
    const float* startT = (const float*)d_in[1];
    const float* endT   = (const float*)d_in[2];
    const float* Wt     = (const float*)d_in[3];
    const float* bt     = (const float*)d_in[4];
    const float* W      = (const float*)d_in[5];
    const float* bias   = (const float*)d_in[6];
    float* z = (float*)d_out;                       // state lives in d_out

    Ctrl*  ctrl    = (Ctrl*)d_ws;
    float* partial = (float*)((char*)d_ws + 64);                  // 1024 floats
    float* Wp      = (float*)((char*)d_ws + 8192);                // 256 KB packed W
    float* zprop   = (float*)((char*)d_ws + 8192 + DIM*DIM*4);    // 16 MB proposed z

    const int copyBlocks = (N_ROWS * DIM / 4) / NT;               // 4096
    node_init_kernel<<<copyBlocks, NT, 0, stream>>>(r, z, ctrl);
    node_pack_w_kernel<<<(DIM * DIM) / NT, NT, 0, stream>>>(W, Wp);

    for (int it = 0; it < NITER; ++it) {
        node_dopri_stage_kernel<<<NBLK, NT, 0, stream>>>(z, startT, endT, Wt, bt, Wp,
                                                         bias, zprop, partial, ctrl);
        node_control_kernel<<<1, 32, 0, stream>>>(ctrl, partial);
        node_commit_kernel<<<copyBlocks, NT, 0, stream>>>(z, zprop, ctrl);
    }
}